// UnMaxPoolWithArgmax_17042430230577
// MI455X (gfx1250) — compile-verified
//
#include <hip/hip_runtime.h>
#include <stdint.h>

// ---------------------------------------------------------------------------
// UnMaxPoolWithArgmax (TF-style), STRIDE=2 along H.
//   B=8, H=512, W=256, C=64  ->  out: (8, 1024, 256, 64) f32
//   out[b, m/(W*C), (m%(W*C))/C, c] += in[b,h,w,c]   (duplicates sum)
//
// Power-of-two shape collapse: C=2^6, W*C=2^14, H*W*C=2^23, 2H*W*C=2^24
//   per-batch output element offset = (m & ~63) | c
//   full output element offset      = (b << 24) + (m & ~63) + c   (< 2^27)
//   byte offset                     = off << 2                    (< 2^29, fits u32)
// => 32-bit unsigned voffset + SGPR base pair: SADDR-form global atomic.
// Memory-bound scatter (~2.3 GB total traffic @ 23.3 TB/s -> ~100 us floor);
// no matrix math, so the CDNA5 wins are NT streaming hints + no-return
// global_atomic_add_f32 (STOREcnt path, no CAS loop).
// ---------------------------------------------------------------------------

// Native vector types (HIP_vector_type wrappers are rejected by the
// __builtin_nontemporal_* builtins).
typedef float              vfloat4  __attribute__((ext_vector_type(4)));
typedef unsigned long long vu64x2   __attribute__((ext_vector_type(2)));

__global__ __launch_bounds__(256) void zero_f4_kernel(float* __restrict__ out,
                                                      unsigned int n) {
    unsigned int t  = blockIdx.x * blockDim.x + threadIdx.x;
    unsigned int i4 = t * 4u;
    if (i4 + 3u < n) {
        vfloat4 z = (vfloat4)0.0f;
        // global_store_b128 ... th:TH_STORE_NT  (streaming, don't pollute caches)
        __builtin_nontemporal_store(z, (vfloat4*)(out + i4));
    } else if (i4 < n) {
        for (unsigned int j = i4; j < n; ++j) out[j] = 0.0f;
    }
}

__device__ __forceinline__ void atomic_add_saddr(float* __restrict__ base,
                                                 unsigned int byte_off,
                                                 float val) {
    // No-return float atomic add, device scope. Tracked by STOREcnt;
    // s_endpgm's implicit wait-idle covers completion. SADDR form:
    //   global_atomic_add_f32 v_off, v_data, s[base:base+1] scope:SCOPE_DEV
    asm volatile("global_atomic_add_f32 %0, %1, %2 scope:SCOPE_DEV"
                 :
                 : "v"(byte_off), "v"(val), "s"(base)
                 : "memory");
}

__global__ __launch_bounds__(256) void unpool_scatter_kernel(
        const float* __restrict__ in,
        const unsigned long long* __restrict__ mask,
        float* __restrict__ out,
        unsigned int n) {
    unsigned int t    = blockIdx.x * blockDim.x + threadIdx.x;
    unsigned int base = t * 4u;

    if (base + 3u < n) {
        // Streaming 128-bit NT loads: 16B of inputs + 32B of masks per thread.
        vfloat4 v  = __builtin_nontemporal_load((const vfloat4*)(in + base));
        vu64x2  mA = __builtin_nontemporal_load((const vu64x2*)(mask + base));
        vu64x2  mB = __builtin_nontemporal_load((const vu64x2*)(mask + base + 2));

        float              vals[4] = { v.x, v.y, v.z, v.w };
        unsigned long long ms[4]   = { mA.x, mA.y, mB.x, mB.y };

#pragma unroll
        for (int k = 0; k < 4; ++k) {
            unsigned int i    = base + (unsigned int)k;
            unsigned int b    = i >> 23;             // batch  (H*W*C = 2^23)
            unsigned int c    = i & 63u;             // channel (C = 64)
            unsigned int moff = ((unsigned int)ms[k]) & ~63u;  // (y,x) part
            unsigned int off  = (b << 24) + moff + c;          // out element, < 2^27
            atomic_add_saddr(out, off << 2, vals[k]);
        }
    } else if (base < n) {
        for (unsigned int i = base; i < n; ++i) {
            float        val  = in[i];
            unsigned int b    = i >> 23;
            unsigned int c    = i & 63u;
            unsigned int moff = ((unsigned int)mask[i]) & ~63u;
            unsigned int off  = (b << 24) + moff + c;
            atomic_add_saddr(out, off << 2, val);
        }
    }
}

extern "C" void kernel_launch(void* const* d_in, const int* in_sizes, int n_in,
                              void* d_out, int out_size, void* d_ws, size_t ws_size,
                              hipStream_t stream) {
    const float*              in   = (const float*)d_in[0];
    const unsigned long long* mask = (const unsigned long long*)d_in[1];
    float*                    out  = (float*)d_out;

    const unsigned int n    = (unsigned int)in_sizes[0];  // 67,108,864
    const unsigned int nout = (unsigned int)out_size;     // 134,217,728

    // Pass 1: zero the output (scatter_nd semantics start from zeros).
    {
        unsigned int threads = (nout + 3u) / 4u;
        unsigned int blocks  = (threads + 255u) / 256u;
        zero_f4_kernel<<<blocks, 256, 0, stream>>>(out, nout);
    }

    // Pass 2: scatter-add with no-return f32 atomics (same stream -> ordered).
    {
        unsigned int threads = (n + 3u) / 4u;
        unsigned int blocks  = (threads + 255u) / 256u;
        unpool_scatter_kernel<<<blocks, 256, 0, stream>>>(in, mask, out, n);
    }
}